// CenterTOpEX_5617817223884
// MI455X (gfx1250) — compile-verified
//
#include <hip/hip_runtime.h>

typedef __attribute__((ext_vector_type(2))) float v2f;
typedef __attribute__((ext_vector_type(8))) float v8f;

#define BATCH 8
#define CH    512
#define NPTS  16384
#define NCLS  2
#define PPB   128            // points per block
#define TPB   256            // threads per block
#define WAVES 8              // 8 waves * 16 points = 128
#define BPB   128            // blocks per batch
#define NBLK  (BATCH*BPB)    // 1024
#define NITER 6

// workspace layout (float offsets)
#define WS_CENTERS 0
#define WS_CN      (WS_CENTERS + BATCH*NCLS*CH)            // 8192
#define WS_PART    (WS_CN + BATCH*NCLS*CH)                 // 16384
#define WS_COUNTS  (WS_PART + NBLK*NCLS*CH)                // +1048576 (ints stored here)
#define WS_DISTS   (WS_COUNTS + NBLK*NCLS)
#define WS_BLOCKMM (WS_DISTS + BATCH*NPTS*NCLS)
#define WS_BMM     (WS_BLOCKMM + NBLK*4)

// output layout (float offsets), concatenated in reference return order
#define O_CENTERS 0
#define O_LABELS  (O_CENTERS + NCLS*CH)          // 1024
#define O_ONEHOT  (O_LABELS + BATCH*NPTS)        // 132096
#define O_WEIGHT  (O_ONEHOT + BATCH*NPTS*NCLS)   // 394240
#define O_LPINIT  (O_WEIGHT + BATCH*NPTS*NCLS)   // 656384
#define O_CINI    (O_LPINIT + BATCH*NPTS)        // 787456

__global__ void k_init_centers(const float* __restrict__ ci, float* __restrict__ ws) {
    int t = blockIdx.x * blockDim.x + threadIdx.x;       // 8192 threads
    ws[WS_CENTERS + t] = ci[t % (NCLS * CH)];
}

__global__ void k_norm_centers(float* __restrict__ ws) {
    int bk = blockIdx.x;                                  // 0..15 = (b,k)
    const float* c = ws + WS_CENTERS + bk * CH;
    float* cn = ws + WS_CN + bk * CH;
    __shared__ float red[TPB];
    int t = threadIdx.x;
    float v0 = c[t], v1 = c[t + 256];
    red[t] = v0 * v0 + v1 * v1;
    __syncthreads();
    for (int s = 128; s > 0; s >>= 1) {
        if (t < s) red[t] += red[t + s];
        __syncthreads();
    }
    float inv = 1.0f / fmaxf(sqrtf(red[0]), 1e-12f);
    cn[t] = v0 * inv;
    cn[t + 256] = v1 * inv;
}

// One full pass over feats: WMMA fp32 dot products vs both normalized centers,
// labels, per-block label-gated channel sums, and (last iter) dists + min/max.
__global__ void k_assign(const float* __restrict__ F, float* __restrict__ ws,
                         float* __restrict__ out, int iter) {
    const int blk = blockIdx.x;
    const int b = blk / BPB;
    const int nbase = (blk % BPB) * PPB;
    const int t = threadIdx.x;
    const int lane = t & 31;
    const int w = t >> 5;
    const int lo = lane & 15;
    const int hi = lane >> 4;

    __shared__ float cnS[NCLS][CH];          // normalized centers
    __shared__ float sS[WAVES][16][2];       // s_k per point
    __shared__ float nsS[TPB];               // per-lane |f|^2 partials
    __shared__ int   labS[PPB];
    __shared__ int   cntS[NCLS];
    __shared__ float redA[TPB], redB[TPB];

    const float* cn = ws + WS_CN + b * NCLS * CH;
    float* cnFlat = &cnS[0][0];
    for (int i = t; i < NCLS * CH; i += TPB) cnFlat[i] = cn[i];
    if (t < NCLS) cntS[t] = 0;
    __syncthreads();

    const float* Fb = F + (size_t)b * CH * NPTS;          // [C][N] layout
    const int n0 = nbase + w * 16;
    // A-matrix lane layout (16x4 f32): lanes 0-15 hold M=lane, K={0,1};
    // lanes 16-31 hold M=lane-16, K={2,3}.
    const float* aptr = Fb + (size_t)(2 * hi) * NPTS + n0 + lo;
    const float bm = (lo < NCLS) ? 1.0f : 0.0f;           // zero unused B columns
    const int ccol = (lo < NCLS) ? lo : 0;

    v8f acc = {0.f, 0.f, 0.f, 0.f, 0.f, 0.f, 0.f, 0.f};
    float ns = 0.0f;
    for (int c0 = 0; c0 < CH; c0 += 4) {
        float ax = aptr[(size_t)c0 * NPTS];               // channel c0 + 2*hi
        float ay = aptr[(size_t)(c0 + 1) * NPTS];         // channel c0 + 1 + 2*hi
        v2f a;  a.x = ax; a.y = ay;
        v2f bb;                                           // B (4x16): row K, col = center
        bb.x = cnS[ccol][c0 + 2 * hi] * bm;
        bb.y = cnS[ccol][c0 + 1 + 2 * hi] * bm;
        acc = __builtin_amdgcn_wmma_f32_16x16x4_f32(
            false, a, false, bb, (short)0, acc, false, false);
        ns = fmaf(ax, ax, ns);
        ns = fmaf(ay, ay, ns);
    }
    nsS[t] = ns;
    // D layout: VGPR r, lanes 0-15 -> (M=r, N=lane); lanes 16-31 -> (M=8+r, N=lane-16)
    if (lo < NCLS) {
#pragma unroll
        for (int r = 0; r < 8; ++r) sS[w][r + 8 * hi][lo] = acc[r];
    }
    __syncthreads();

    float d0 = 0.0f, d1 = 0.0f;
    if (t < PPB) {
        int wv = t >> 4, m = t & 15;
        float s0 = sS[wv][m][0];
        float s1 = sS[wv][m][1];
        int lbl = (s1 > s0) ? 1 : 0;                      // argmin dist, ties -> 0
        labS[t] = lbl;
        atomicAdd(&cntS[lbl], 1);
        int gn = b * NPTS + nbase + t;
        if (iter == 0) out[O_LPINIT + gn] = (float)lbl;
        if (iter == NITER - 1) {
            float nf2 = nsS[wv * 32 + m] + nsS[wv * 32 + m + 16];
            float inv = 1.0f / fmaxf(sqrtf(nf2), 1e-12f);
            d0 = 0.5f * (1.0f - s0 * inv);
            d1 = 0.5f * (1.0f - s1 * inv);
            out[O_LABELS + gn] = (float)lbl;
            out[O_ONEHOT + gn * 2 + 0] = (lbl == 0) ? 1.0f : 0.0f;
            out[O_ONEHOT + gn * 2 + 1] = (lbl == 1) ? 1.0f : 0.0f;
            ws[WS_DISTS + gn * 2 + 0] = d0;
            ws[WS_DISTS + gn * 2 + 1] = d1;
        }
    }
    if (iter == NITER - 1) {                              // block min/max of dists
        redA[t] = (t < PPB) ? d0 : __builtin_inff();
        redB[t] = (t < PPB) ? d1 : __builtin_inff();
        __syncthreads();
        for (int s = 128; s > 0; s >>= 1) {
            if (t < s) { redA[t] = fminf(redA[t], redA[t + s]); redB[t] = fminf(redB[t], redB[t + s]); }
            __syncthreads();
        }
        if (t == 0) { ws[WS_BLOCKMM + blk * 4 + 0] = redA[0]; ws[WS_BLOCKMM + blk * 4 + 2] = redB[0]; }
        __syncthreads();
        redA[t] = (t < PPB) ? d0 : -__builtin_inff();
        redB[t] = (t < PPB) ? d1 : -__builtin_inff();
        __syncthreads();
        for (int s = 128; s > 0; s >>= 1) {
            if (t < s) { redA[t] = fmaxf(redA[t], redA[t + s]); redB[t] = fmaxf(redB[t], redB[t + s]); }
            __syncthreads();
        }
        if (t == 0) { ws[WS_BLOCKMM + blk * 4 + 1] = redA[0]; ws[WS_BLOCKMM + blk * 4 + 3] = redB[0]; }
    }
    __syncthreads();

    // Phase B: deterministic per-block label-gated channel sums (reads hit L2).
#pragma unroll
    for (int half = 0; half < 2; ++half) {
        int cc = t + half * TPB;
        const float* row = Fb + (size_t)cc * NPTS + nbase;
        float s0 = 0.0f, s1 = 0.0f;
        for (int n = 0; n < PPB; ++n) {
            float v = row[n];
            if (labS[n]) s1 += v; else s0 += v;
        }
        ws[WS_PART + (size_t)blk * NCLS * CH + 0 * CH + cc] = s0;
        ws[WS_PART + (size_t)blk * NCLS * CH + 1 * CH + cc] = s1;
    }
    if (t < NCLS) ((int*)(ws + WS_COUNTS))[blk * NCLS + t] = cntS[t];
}

__global__ void k_reduce_centers(float* __restrict__ ws) {
    int tid = blockIdx.x * blockDim.x + threadIdx.x;      // 8192 = B*K*C
    int b = tid / (NCLS * CH);
    int k = (tid / CH) % NCLS;
    int c = tid % CH;
    const float* part = ws + WS_PART;
    const int* cnts = (const int*)(ws + WS_COUNTS);
    float s = 0.0f; int cnt = 0;
    for (int j = 0; j < BPB; ++j) {                       // fixed order -> deterministic
        int blk = b * BPB + j;
        s += part[(size_t)blk * NCLS * CH + k * CH + c];
        cnt += cnts[blk * NCLS + k];
    }
    ws[WS_CENTERS + tid] = s / ((float)cnt + 1.0f);
}

__global__ void k_cinidist(const float* __restrict__ ci, const float* __restrict__ ws,
                           float* __restrict__ out) {
    __shared__ float rd[TPB], ra[TPB], rb[TPB];
    int t = threadIdx.x;
    float total = 0.0f;
    for (int bk = 0; bk < BATCH * NCLS; ++bk) {
        const float* nc = ws + WS_CENTERS + bk * CH;      // centers after iter-0 update
        const float* ic = ci + (bk % NCLS) * CH;          // centerInit
        float d = 0.f, a2 = 0.f, b2 = 0.f;
        for (int c = t; c < CH; c += TPB) {
            float x = nc[c], y = ic[c];
            d += x * y; a2 += x * x; b2 += y * y;
        }
        rd[t] = d; ra[t] = a2; rb[t] = b2;
        __syncthreads();
        for (int s = 128; s > 0; s >>= 1) {
            if (t < s) { rd[t] += rd[t + s]; ra[t] += ra[t + s]; rb[t] += rb[t + s]; }
            __syncthreads();
        }
        total += rd[0] / (fmaxf(sqrtf(ra[0]), 1e-8f) * fmaxf(sqrtf(rb[0]), 1e-8f));
        __syncthreads();
    }
    if (t == 0) out[O_CINI] = total / (float)(BATCH * NCLS);
}

__global__ void k_minmax(float* __restrict__ ws) {
    int t = threadIdx.x;
    if (t >= BATCH * NCLS) return;
    int b = t / NCLS, k = t % NCLS;
    float mn = __builtin_inff(), mx = -__builtin_inff();
    for (int j = 0; j < BPB; ++j) {
        int blk = b * BPB + j;
        mn = fminf(mn, ws[WS_BLOCKMM + blk * 4 + k * 2 + 0]);
        mx = fmaxf(mx, ws[WS_BLOCKMM + blk * 4 + k * 2 + 1]);
    }
    ws[WS_BMM + t * 2 + 0] = mn;
    ws[WS_BMM + t * 2 + 1] = mx;
}

__global__ void k_weight(const float* __restrict__ ws, float* __restrict__ out) {
    int idx = blockIdx.x * blockDim.x + threadIdx.x;      // B*N*K
    int b = idx / (NPTS * NCLS);
    int k = idx % NCLS;
    float d = ws[WS_DISTS + idx];
    float mn = ws[WS_BMM + (b * NCLS + k) * 2 + 0];
    float mx = ws[WS_BMM + (b * NCLS + k) * 2 + 1];
    out[O_WEIGHT + idx] = 1.0f - (d - mn) / (mx - mn + 1e-7f);
}

__global__ void k_centers_mean(const float* __restrict__ ws, float* __restrict__ out) {
    int r = blockIdx.x * blockDim.x + threadIdx.x;        // 1024 = K*C
    float s = 0.0f;
    for (int b = 0; b < BATCH; ++b) s += ws[WS_CENTERS + b * NCLS * CH + r];
    out[O_CENTERS + r] = s * (1.0f / (float)BATCH);
}

extern "C" void kernel_launch(void* const* d_in, const int* in_sizes, int n_in,
                              void* d_out, int out_size, void* d_ws, size_t ws_size,
                              hipStream_t stream) {
    (void)in_sizes; (void)n_in; (void)out_size; (void)ws_size;
    const float* F  = (const float*)d_in[0];   // FeatureT [8,512,128,128] fp32
    const float* ci = (const float*)d_in[1];   // centerInit [2,512] fp32
    float* out = (float*)d_out;
    float* ws  = (float*)d_ws;

    k_init_centers<<<32, TPB, 0, stream>>>(ci, ws);
    for (int i = 0; i < NITER; ++i) {
        k_norm_centers<<<BATCH * NCLS, TPB, 0, stream>>>(ws);
        k_assign<<<NBLK, TPB, 0, stream>>>(F, ws, out, i);
        k_reduce_centers<<<32, TPB, 0, stream>>>(ws);
        if (i == 0) k_cinidist<<<1, TPB, 0, stream>>>(ci, ws, out);
    }
    k_minmax<<<1, 32, 0, stream>>>(ws);
    k_weight<<<(BATCH * NPTS * NCLS) / TPB, TPB, 0, stream>>>(ws, out);
    k_centers_mean<<<(NCLS * CH) / TPB, TPB, 0, stream>>>(ws, out);
}